// SpectraCellType_13967233647508
// MI455X (gfx1250) — compile-verified
//
#include <hip/hip_runtime.h>
#include <math.h>

#define DEV_INLINE __device__ __forceinline__

typedef float v2f __attribute__((ext_vector_type(2)));
typedef float v8f __attribute__((ext_vector_type(8)));

// ---- problem constants ----
constexpr int C_  = 3;
constexpr int P_  = 4000;
constexpr int LG_ = 30;
constexpr int LC_ = 20;
constexpr int NC_ = 1500;
constexpr int K_  = LG_ + LC_;   // 50
constexpr float LAM   = 0.01f;
constexpr float DELTA = 0.001f;
constexpr float FRAC  = (float)NC_ / (float)(C_ * NC_); // 1/3
constexpr float LN2   = 0.69314718055994531f;

// padded strides (multiples of 4 for WMMA K-chunks, even for float2 loads)
constexpr int LGP = 32;  // 30 -> 32
constexpr int KP  = 52;  // 50 -> 52

constexpr int TP  = P_ / 16;   // 250 column tiles along P
constexpr int TPH = TP / 2;    // 125 column tile PAIRS (16x32 strip per wave)

constexpr int CT_PAIRS = C_ * TP * TPH;        // 93750
constexpr int G_PAIRS  = TP * TPH;             // 31250
constexpr int GRAPH_PAIRS = CT_PAIRS + G_PAIRS;

constexpr int RT1      = (NC_ + 15) / 16;      // 94 row tiles for NC
constexpr int T1_PAIRS = C_ * RT1 * TPH;       // 35250

constexpr int BLK = 256;                       // 8 wave32 per block
constexpr int GRAPH_BLOCKS = 1024;
constexpr int T1_BLOCKS    = 512;

// ---- workspace layout (float offsets, all even => 8B aligned) ----
constexpr size_t OFF_TG   = 0;                                   // [P][32]
constexpr size_t OFF_UG   = OFF_TG + (size_t)P_ * LGP;           // [P][32]
constexpr size_t OFF_TC   = OFF_UG + (size_t)P_ * LGP;           // [C*P][20]
constexpr size_t OFF_UC   = OFF_TC + (size_t)C_ * P_ * LC_;      // [C*P][20]
constexpr size_t OFF_TF   = OFF_UC + (size_t)C_ * P_ * LC_;      // [C*P][52]
constexpr size_t OFF_AE   = OFF_TF + (size_t)C_ * P_ * KP;       // [C*NC][52]
constexpr size_t OFF_PART = OFF_AE + (size_t)C_ * NC_ * KP;      // partials

DEV_INLINE float sigm(float x) { return 1.0f / (1.0f + __expf(-x)); }

// raw v_log_f32 (log base 2); args here are always normal & > 0
DEV_INLINE float log2_raw(float x) { return __builtin_amdgcn_logf(x); }

DEV_INLINE v8f wmma4(v2f a, v2f b, v8f c) {
    // V_WMMA_F32_16X16X4_F32 : D[16x16] = A[16x4] * B[4x16] + C
    return __builtin_amdgcn_wmma_f32_16x16x4_f32(
        /*neg_a=*/false, a, /*neg_b=*/false, b,
        /*c_mod=*/(short)0, c, /*reuse_a=*/false, /*reuse_b=*/false);
}

// ---------------------------------------------------------------------------
// Kernel 1: transforms (softmax tg/tc, scaled theta_full, exp(alpha))
// ---------------------------------------------------------------------------
__global__ void k_transform(const float* __restrict__ theta_g,
                            const float* __restrict__ gs_g,
                            const float* __restrict__ theta_c,
                            const float* __restrict__ gs_c,
                            const float* __restrict__ alpha_c,
                            float* __restrict__ ws) {
    int idx = blockIdx.x * blockDim.x + threadIdx.x;
    float* tg = ws + OFF_TG;
    float* tc = ws + OFF_TC;
    float* tf = ws + OFF_TF;
    float* ae = ws + OFF_AE;

    if (idx < P_) {
        // global softmax row p over LG
        int p = idx;
        const float* th = theta_g + (size_t)p * LG_;
        float mx = -1e30f;
        for (int l = 0; l < LG_; ++l) mx = fmaxf(mx, th[l]);
        float sum = 0.f;
        for (int l = 0; l < LG_; ++l) sum += __expf(th[l] - mx);
        float inv = 1.0f / sum;
        float gsg = sigm(gs_g[p]) + DELTA;
        for (int l = 0; l < LG_; ++l) {
            float v = __expf(th[l] - mx) * inv;
            tg[(size_t)p * LGP + l] = v;
            float vs = v * gsg;
            for (int c = 0; c < C_; ++c)
                tf[((size_t)c * P_ + p) * KP + l] = vs;
        }
        tg[(size_t)p * LGP + 30] = 0.f;
        tg[(size_t)p * LGP + 31] = 0.f;
    } else if (idx < P_ + C_ * P_) {
        // per-cell-type softmax row cp over LC
        int cp = idx - P_;
        const float* th = theta_c + (size_t)cp * LC_;
        float mx = -1e30f;
        for (int l = 0; l < LC_; ++l) mx = fmaxf(mx, th[l]);
        float sum = 0.f;
        for (int l = 0; l < LC_; ++l) sum += __expf(th[l] - mx);
        float inv = 1.0f / sum;
        float gsc = sigm(gs_c[cp]) + DELTA;
        for (int l = 0; l < LC_; ++l) {
            float v = __expf(th[l] - mx) * inv;
            tc[(size_t)cp * LC_ + l] = v;
            tf[(size_t)cp * KP + LG_ + l] = v * gsc;
        }
        tf[(size_t)cp * KP + 50] = 0.f;
        tf[(size_t)cp * KP + 51] = 0.f;
    } else {
        // alpha = exp(alpha_c), padded to stride KP with zeros
        int e = idx - (P_ + C_ * P_);
        if (e < C_ * NC_ * KP) {
            int row = e / KP, k = e % KP;
            ae[(size_t)row * KP + k] =
                (k < K_) ? __expf(alpha_c[(size_t)row * K_ + k]) : 0.f;
        }
    }
}

// ---------------------------------------------------------------------------
// Kernel 2: U = t * e  (e = symmetrized sigmoid of eta, computed on the fly)
// ---------------------------------------------------------------------------
__global__ void k_umat(const float* __restrict__ eta_g,
                       const float* __restrict__ eta_c,
                       float* __restrict__ ws) {
    int idx = blockIdx.x * blockDim.x + threadIdx.x;
    const float* tg = ws + OFF_TG;
    float* Ug = ws + OFF_UG;
    const float* tc = ws + OFF_TC;
    float* Uc = ws + OFF_UC;

    if (idx < P_ * LGP) {
        int p = idx / LGP, m = idx % LGP;
        float s = 0.f;
        if (m < LG_) {
            for (int l = 0; l < LG_; ++l) {
                float e = 0.5f * (sigm(eta_g[l * LG_ + m]) + sigm(eta_g[m * LG_ + l]));
                s += tg[(size_t)p * LGP + l] * e;
            }
        }
        Ug[idx] = s;
    } else {
        int j = idx - P_ * LGP;
        if (j < C_ * P_ * LC_) {
            int cp = j / LC_, m = j % LC_;
            int c = cp / P_;
            const float* ec = eta_c + (size_t)c * LC_ * LC_;
            float s = 0.f;
            for (int l = 0; l < LC_; ++l) {
                float e = 0.5f * (sigm(ec[l * LC_ + m]) + sigm(ec[m * LC_ + l]));
                s += tc[(size_t)cp * LC_ + l] * e;
            }
            Uc[j] = s;
        }
    }
}

// ---------------------------------------------------------------------------
// Kernel 3: fused graph terms (ct + global). Each wave produces a 16x32 strip
// of `mat` via WMMA (A reused across the two column tiles), then consumes the
// full 128B cachelines of adj/weights branchlessly with raw v_log_f32.
// All graph contributions are of the form  -x*log(arg),  so we accumulate
// x*log2(arg) and scale once by ln2.
// ---------------------------------------------------------------------------
__global__ void k_graph(const float* __restrict__ adj_ct,
                        const float* __restrict__ w_ct,
                        const float* __restrict__ adj_g,
                        const float* __restrict__ w_g,
                        const float* __restrict__ kappa_c,
                        const float* __restrict__ rho_c,
                        const float* __restrict__ kappa_g,
                        const float* __restrict__ rho_g,
                        const float* __restrict__ ws,
                        float* __restrict__ partial) {
    const float* tg = ws + OFF_TG;
    const float* Ug = ws + OFF_UG;
    const float* tc = ws + OFF_TC;
    const float* Uc = ws + OFF_UC;

    int lane = threadIdx.x & 31;
    int m15 = lane & 15, hi = lane >> 4;
    int wid = blockIdx.x * (BLK / 32) + (threadIdx.x >> 5);
    int nw  = gridDim.x * (BLK / 32);

    float kg = sigm(kappa_g[0]), rg = sigm(rho_g[0]);
    // arg2 = g2a*m + g2b ; arg3 = g3c - g3a*m
    float g2a = (1.f - rg) * (1.f - kg), g2b = (1.f - rg) * kg;
    float g3a = (1.f - kg) * (1.f - rg), g3c = g3a + rg;

    float s = 0.f;

    for (int t = wid; t < GRAPH_PAIRS; t += nw) {
        if (t < CT_PAIRS) {
            int c  = t / (TP * TPH);
            int r  = t % (TP * TPH);
            int ti = r / TPH, tk0 = (r % TPH) * 2;
            const float* A  = Uc + ((size_t)c * P_ + ti * 16 + m15) * LC_ + 2 * hi;
            const float* B0 = tc + ((size_t)c * P_ + tk0 * 16 + m15) * LC_ + 2 * hi;
            v8f acc0 = {0.f, 0.f, 0.f, 0.f, 0.f, 0.f, 0.f, 0.f};
            v8f acc1 = acc0;
            #pragma unroll
            for (int kk = 0; kk < LC_; kk += 4) {
                v2f av = *(const v2f*)(A + kk);
                acc0 = wmma4(av, *(const v2f*)(B0 + kk), acc0);
                acc1 = wmma4(av, *(const v2f*)(B0 + 16 * LC_ + kk), acc1);
            }

            float kc = sigm(kappa_c[c]), rc = sigm(rho_c[c]);
            float c2a = (1.f - rc) * (1.f - kc), c2b = (1.f - rc) * kc;
            float c3a = (1.f - kc) * (1.f - rc), c3c = c3a + rc;

            int kcol = tk0 * 16 + m15;
            float ls = 0.f;  // accumulates x*log2(arg)
            #pragma unroll
            for (int v = 0; v < 8; ++v) {
                int i = ti * 16 + v + 8 * hi;
                size_t gidx = ((size_t)c * P_ + i) * P_ + kcol;
                float a0 = adj_ct[gidx];
                float w0 = w_ct[gidx];
                float a1 = adj_ct[gidx + 16];
                float w1 = w_ct[gidx + 16];
                float m0 = acc0[v], m1 = acc1[v];
                ls += (a0 * w0) * log2_raw(c2a * m0 + c2b);
                ls += (a1 * w1) * log2_raw(c2a * m1 + c2b);
                float x30 = (i != kcol)      ? (1.f - a0) : 0.f;
                float x31 = (i != kcol + 16) ? (1.f - a1) : 0.f;
                ls += x30 * log2_raw(c3c - c3a * m0);
                ls += x31 * log2_raw(c3c - c3a * m1);
            }
            s -= (FRAC * LN2) * ls;
        } else {
            int r  = t - CT_PAIRS;
            int ti = r / TPH, tk0 = (r % TPH) * 2;
            const float* A  = Ug + ((size_t)(ti * 16 + m15)) * LGP + 2 * hi;
            const float* B0 = tg + ((size_t)(tk0 * 16 + m15)) * LGP + 2 * hi;
            v8f acc0 = {0.f, 0.f, 0.f, 0.f, 0.f, 0.f, 0.f, 0.f};
            v8f acc1 = acc0;
            #pragma unroll
            for (int kk = 0; kk < LGP; kk += 4) {
                v2f av = *(const v2f*)(A + kk);
                acc0 = wmma4(av, *(const v2f*)(B0 + kk), acc0);
                acc1 = wmma4(av, *(const v2f*)(B0 + 16 * LGP + kk), acc1);
            }

            int kcol = tk0 * 16 + m15;
            float ls = 0.f;
            #pragma unroll
            for (int v = 0; v < 8; ++v) {
                int i = ti * 16 + v + 8 * hi;
                size_t gidx = (size_t)i * P_ + kcol;
                float a0 = adj_g[gidx];
                float w0 = w_g[gidx];
                float a1 = adj_g[gidx + 16];
                float w1 = w_g[gidx + 16];
                float m0 = acc0[v], m1 = acc1[v];
                ls += (a0 * w0) * log2_raw(g2a * m0 + g2b);
                ls += (a1 * w1) * log2_raw(g2a * m1 + g2b);
                float x30 = (i != kcol)      ? (1.f - a0) : 0.f;
                float x31 = (i != kcol + 16) ? (1.f - a1) : 0.f;
                ls += x30 * log2_raw(g3c - g3a * m0);
                ls += x31 * log2_raw(g3c - g3a * m1);
            }
            s -= LN2 * ls;
        }
    }

    __shared__ float red[BLK];
    red[threadIdx.x] = s;
    __syncthreads();
    for (int off = BLK / 2; off > 0; off >>= 1) {
        if (threadIdx.x < off) red[threadIdx.x] += red[threadIdx.x + off];
        __syncthreads();
    }
    if (threadIdx.x == 0) partial[blockIdx.x] = red[0];
}

// ---------------------------------------------------------------------------
// Kernel 4: fused term1: 16x32 recon strip via WMMA, then
//           sum(recon) - ln2 * sum(X*log2(recon))
// ---------------------------------------------------------------------------
__global__ void k_term1(const float* __restrict__ X,
                        const float* __restrict__ ws,
                        float* __restrict__ partial) {
    const float* ae = ws + OFF_AE;
    const float* tf = ws + OFF_TF;

    int lane = threadIdx.x & 31;
    int m15 = lane & 15, hi = lane >> 4;
    int wid = blockIdx.x * (BLK / 32) + (threadIdx.x >> 5);
    int nw  = gridDim.x * (BLK / 32);

    float s = 0.f;
    for (int t = wid; t < T1_PAIRS; t += nw) {
        int c  = t / (RT1 * TPH);
        int r  = t % (RT1 * TPH);
        int ti = r / TPH, tk0 = (r % TPH) * 2;

        int rowA = ti * 16 + m15;
        bool okA = rowA < NC_;
        const float* A  = ae + ((size_t)c * NC_ + (okA ? rowA : 0)) * KP + 2 * hi;
        const float* B0 = tf + ((size_t)c * P_ + tk0 * 16 + m15) * KP + 2 * hi;

        v8f acc0 = {0.f, 0.f, 0.f, 0.f, 0.f, 0.f, 0.f, 0.f};
        v8f acc1 = acc0;
        #pragma unroll
        for (int kk = 0; kk < KP; kk += 4) {
            v2f av;
            if (okA) av = *(const v2f*)(A + kk);
            else { av.x = 0.f; av.y = 0.f; }
            acc0 = wmma4(av, *(const v2f*)(B0 + kk), acc0);
            acc1 = wmma4(av, *(const v2f*)(B0 + 16 * KP + kk), acc1);
        }

        int kcol = tk0 * 16 + m15;
        float lsm = 0.f;  // sum(recon)
        float lsl = 0.f;  // sum(X * log2(recon))
        #pragma unroll
        for (int v = 0; v < 8; ++v) {
            int i = ti * 16 + v + 8 * hi;
            if (i < NC_) {
                size_t gidx = ((size_t)c * NC_ + i) * P_ + kcol;
                float x0 = X[gidx];
                float x1 = X[gidx + 16];
                float m0 = acc0[v], m1 = acc1[v];
                lsm += m0 + m1;
                lsl += x0 * log2_raw(m0);
                lsl += x1 * log2_raw(m1);
            }
        }
        s += LAM * (lsm - LN2 * lsl);
    }

    __shared__ float red[BLK];
    red[threadIdx.x] = s;
    __syncthreads();
    for (int off = BLK / 2; off > 0; off >>= 1) {
        if (threadIdx.x < off) red[threadIdx.x] += red[threadIdx.x + off];
        __syncthreads();
    }
    if (threadIdx.x == 0) partial[GRAPH_BLOCKS + blockIdx.x] = red[0];
}

// ---------------------------------------------------------------------------
// Kernel 5: deterministic final reduction of all partials
// ---------------------------------------------------------------------------
__global__ void k_reduce(const float* __restrict__ partial,
                         float* __restrict__ out) {
    __shared__ float red[BLK];
    float s = 0.f;
    for (int i = threadIdx.x; i < GRAPH_BLOCKS + T1_BLOCKS; i += BLK)
        s += partial[i];
    red[threadIdx.x] = s;
    __syncthreads();
    for (int off = BLK / 2; off > 0; off >>= 1) {
        if (threadIdx.x < off) red[threadIdx.x] += red[threadIdx.x + off];
        __syncthreads();
    }
    if (threadIdx.x == 0) out[0] = red[0];
}

// ---------------------------------------------------------------------------
extern "C" void kernel_launch(void* const* d_in, const int* in_sizes, int n_in,
                              void* d_out, int out_size, void* d_ws, size_t ws_size,
                              hipStream_t stream) {
    (void)in_sizes; (void)n_in; (void)out_size; (void)ws_size;

    const float* X        = (const float*)d_in[0];
    const float* adj_ct   = (const float*)d_in[1];
    const float* w_ct     = (const float*)d_in[2];
    const float* adj_g    = (const float*)d_in[3];
    const float* w_g      = (const float*)d_in[4];
    const float* theta_g  = (const float*)d_in[5];
    const float* eta_g    = (const float*)d_in[6];
    const float* gs_g     = (const float*)d_in[7];
    const float* kappa_g  = (const float*)d_in[8];
    const float* rho_g    = (const float*)d_in[9];
    const float* theta_c  = (const float*)d_in[10];
    const float* eta_c    = (const float*)d_in[11];
    const float* gs_c     = (const float*)d_in[12];
    const float* alpha_c  = (const float*)d_in[13];
    const float* kappa_c  = (const float*)d_in[14];
    const float* rho_c    = (const float*)d_in[15];

    float* ws  = (float*)d_ws;
    float* out = (float*)d_out;

    int n1 = P_ + C_ * P_ + C_ * NC_ * KP;
    k_transform<<<(n1 + BLK - 1) / BLK, BLK, 0, stream>>>(
        theta_g, gs_g, theta_c, gs_c, alpha_c, ws);

    int n2 = P_ * LGP + C_ * P_ * LC_;
    k_umat<<<(n2 + BLK - 1) / BLK, BLK, 0, stream>>>(eta_g, eta_c, ws);

    k_graph<<<GRAPH_BLOCKS, BLK, 0, stream>>>(
        adj_ct, w_ct, adj_g, w_g, kappa_c, rho_c, kappa_g, rho_g,
        ws, ws + OFF_PART);

    k_term1<<<T1_BLOCKS, BLK, 0, stream>>>(X, ws, ws + OFF_PART);

    k_reduce<<<1, BLK, 0, stream>>>(ws + OFF_PART, out);
}